// EarthMoversDistanceLoss_77824807403971
// MI455X (gfx1250) — compile-verified
//
#include <hip/hip_runtime.h>
#include <hip/hip_bf16.h>

// EMD loss: per row b, sum_j (cumsum(x)[b,j] - cumsum(y)[b,j])^2
// = sum_j (cumsum(x-y)[b,j])^2.
// Bandwidth-bound: 256 MiB in / 0.5 MiB out -> ~11.5 us floor @ 23.3 TB/s.
// Scan implemented as D x U (upper-triangular ones) via V_WMMA_F32_16X16X4_F32,
// which is exactly the original masked-matmul formulation.
//
// Prefetch note: locality 3 lowers to SE-scope global_prefetch_b8 (fills GL2).
// That is the *right* scope here: with ~32 streaming waves per WGP, the 64 KB
// WGP$ can't hold prefetched lines until use, but GL2 (192 MB) can. WGP-scope
// prefetch via inline asm was tried and rejected: the volatile asm fenced the
// scheduler and collapsed the 8-deep load pipeline to depth 1.

typedef float v2f __attribute__((ext_vector_type(2)));
typedef float v4f __attribute__((ext_vector_type(4)));
typedef float v8f __attribute__((ext_vector_type(8)));

#define EMD_BINS 256

__global__ __launch_bounds__(256) void emd_wmma_kernel(
    const float* __restrict__ x, const float* __restrict__ y,
    float* __restrict__ out, int num_tiles) {
  const int lane = threadIdx.x & 31;
  const int wave = threadIdx.x >> 5;
  const int tile = blockIdx.x * 8 + wave;   // uniform per wave
  if (tile >= num_tiles) return;            // whole-wave exit: EXEC all-1 for WMMA
  const int r0 = tile << 4;                 // first row of this 16-row tile

  const int n = lane & 15;                  // output column / data row within tile
  const int h = lane >> 4;                  // lane half (selects K offset in A/B layout)

  // B operand = 16x16 upper-triangular ones, U[k][nn] = (nn >= k), built once.
  // Sub-step j covers K rows 4j..4j+3. Layout symmetric to documented 32-bit A
  // layout: v0 holds row K=4j+2h, v1 holds row K=4j+1+2h, column = n.
  v2f B[4];
#pragma unroll
  for (int j = 0; j < 4; ++j) {
    const int k0 = 4 * j + 2 * h;
    B[j].x = (n >= k0)     ? 1.0f : 0.0f;
    B[j].y = (n >= k0 + 1) ? 1.0f : 0.0f;
  }

  // Documented 32-bit A 16x4 layout: lane holds M=n, v0 = K=(0|2), v1 = K=(1|3)
  // -> this lane always reads float2 at row (r0+n), col (c*16 + 4j + 2h).
  const float* xrow = x + (size_t)(r0 + n) * EMD_BINS + 2 * h;
  const float* yrow = y + (size_t)(r0 + n) * EMD_BINS + 2 * h;

  float off[8], acc[8];                     // per-lane: rows i+8h (C-matrix layout)
#pragma unroll
  for (int i = 0; i < 8; ++i) { off[i] = 0.0f; acc[i] = 0.0f; }

#pragma unroll 4
  for (int c = 0; c < 16; ++c) {            // 16 chunks of 16 bins
    if (c + 2 < 16) {                       // pull two chunks (128 B) ahead into GL2
      __builtin_prefetch(xrow + (c + 2) * 16, 0, 3);  // -> SE-scope prefetch
      __builtin_prefetch(yrow + (c + 2) * 16, 0, 3);
    }
    v8f s = {0.f, 0.f, 0.f, 0.f, 0.f, 0.f, 0.f, 0.f};
#pragma unroll
    for (int j = 0; j < 4; ++j) {           // K=16 as 4 accumulating 16x16x4 WMMAs
      const int col = c * 16 + 4 * j;
      v2f ax = *(const v2f*)(xrow + col);
      v2f ay = *(const v2f*)(yrow + col);
      v2f a;
      a.x = ax.x - ay.x;
      a.y = ax.y - ay.y;
      s = __builtin_amdgcn_wmma_f32_16x16x4_f32(
          /*neg_a=*/false, a, /*neg_b=*/false, B[j],
          /*c_mod=*/(short)0, s, /*reuse_a=*/false, /*reuse_b=*/false);
    }
    // C layout: s[i] = within-chunk cdf of row (i+8h) at bin n of this chunk.
    // Lane 15 (h=0) / lane 31 (h=1) hold the chunk-final cdf -> next offset.
#pragma unroll
    for (int i = 0; i < 8; ++i) {
      float v = s[i] + off[i];              // global cdf value at this bin
      acc[i] = fmaf(v, v, acc[i]);
      off[i] = __shfl(v, 15 + 16 * h, 32);  // ds_bpermute broadcast of running total
    }
  }

  // Reduce squared-error over the 16 bins held across lanes of each half.
#pragma unroll
  for (int i = 0; i < 8; ++i) {
    float a = acc[i];
    a += __shfl_xor(a, 1, 32);
    a += __shfl_xor(a, 2, 32);
    a += __shfl_xor(a, 4, 32);
    a += __shfl_xor(a, 8, 32);
    acc[i] = a;
  }

  // Lane 0 holds rows r0..r0+7, lane 16 holds rows r0+8..r0+15.
  if (n == 0) {
    float* dst = out + r0 + 8 * h;
    v4f lo = {acc[0], acc[1], acc[2], acc[3]};
    v4f hi = {acc[4], acc[5], acc[6], acc[7]};
    *(v4f*)(dst)     = lo;                  // global_store_b128
    *(v4f*)(dst + 4) = hi;
  }
}

extern "C" void kernel_launch(void* const* d_in, const int* in_sizes, int n_in,
                              void* d_out, int out_size, void* d_ws, size_t ws_size,
                              hipStream_t stream) {
  const float* x = (const float*)d_in[0];
  const float* y = (const float*)d_in[1];
  float* out = (float*)d_out;

  const int rows = in_sizes[0] / EMD_BINS;  // 131072
  const int num_tiles = rows / 16;          // 8192 (N divisible by 16)
  const int blocks = (num_tiles + 7) / 8;   // 8 waves (tiles) per 256-thread block

  emd_wmma_kernel<<<blocks, 256, 0, stream>>>(x, y, out, num_tiles);
}